// SepGPDLayer_80135499809097
// MI455X (gfx1250) — compile-verified
//
#include <hip/hip_runtime.h>

typedef float v2f __attribute__((ext_vector_type(2)));
typedef float v8f __attribute__((ext_vector_type(8)));

#define IN_C 31
#define TOT  44          // 8 mean + 8 scale + 28 angles
#define HW   65536       // 256*256
#define NPIX (4 * HW)    // B*H*W = 262144
#define M_ELEMS (NPIX * 8)

__global__ __launch_bounds__(256) void sepgpd_kernel(
    const float* __restrict__ x,      // (4,31,256,256)
    const float* __restrict__ Wenc,   // (44,31)
    const float* __restrict__ benc,   // (44,)
    float* __restrict__ out)          // m (NPIX*8) ++ S (NPIX*64)
{
  // Wp: padded weights, layout [o][k], o<48, k<32; k==31 holds bias (A supplies 1.0)
  __shared__ float Wp[48 * 32];
  // per-wave w-tile staging: 8 waves * 32 pixels * 48 channels
  __shared__ float wst[8 * 32 * 48];

  const int tid = threadIdx.x;
  for (int i = tid; i < 48 * 32; i += 256) {
    int o = i >> 5, k = i & 31;
    float v = 0.0f;
    if (o < TOT) v = (k < IN_C) ? Wenc[o * IN_C + k] : benc[o];
    Wp[i] = v;
  }
  __syncthreads();

  const int lane = tid & 31;
  const int wave = tid >> 5;
  const int waveBase = blockIdx.x * 256 + wave * 32;  // first global pixel of this wave
  const int b = waveBase >> 16;                       // batch
  const int q = waveBase & (HW - 1);                  // pixel within batch
  const float* xb = x + (size_t)b * IN_C * HW + q;

  const int mrow  = lane & 15;   // M (pixel) / N (channel) index within tile
  const int khalf = lane >> 4;   // which K half-pair this lane holds

  // ---- A fragments: 2 M-tiles of 16 pixels, K padded 31->32 (k==31 -> 1.0 for bias) ----
  v2f A0[8], A1[8];
#pragma unroll
  for (int kk = 0; kk < 8; ++kk) {
#pragma unroll
    for (int r = 0; r < 2; ++r) {
      int k = 4 * kk + 2 * khalf + r;
      A0[kk][r] = (k < IN_C) ? xb[k * HW + mrow]      : 1.0f;
      A1[kk][r] = (k < IN_C) ? xb[k * HW + 16 + mrow] : 1.0f;
    }
  }

  v8f acc0[3], acc1[3];
#pragma unroll
  for (int t = 0; t < 3; ++t) {
    acc0[t] = (v8f){0, 0, 0, 0, 0, 0, 0, 0};
    acc1[t] = (v8f){0, 0, 0, 0, 0, 0, 0, 0};
  }

  // ---- GEMM: 8 K-steps x 3 N-tiles x 2 M-tiles = 48 v_wmma_f32_16x16x4_f32 ----
#pragma unroll
  for (int kk = 0; kk < 8; ++kk) {
#pragma unroll
    for (int t = 0; t < 3; ++t) {
      v2f Bf = *(const v2f*)&Wp[(16 * t + mrow) * 32 + 4 * kk + 2 * khalf];
      acc0[t] = __builtin_amdgcn_wmma_f32_16x16x4_f32(
          false, A0[kk], false, Bf, (short)0, acc0[t], false, false);
      acc1[t] = __builtin_amdgcn_wmma_f32_16x16x4_f32(
          false, A1[kk], false, Bf, (short)0, acc1[t], false, false);
    }
  }

  // ---- transpose w-tile through LDS: lane holds channel -> lane holds pixel ----
  float* slot = &wst[wave * 32 * 48];
  const int prow = khalf ? 8 : 0;
#pragma unroll
  for (int t = 0; t < 3; ++t) {
#pragma unroll
    for (int v = 0; v < 8; ++v) {
      slot[(prow + v) * 48 + 16 * t + mrow]      = acc0[t][v];
      slot[(16 + prow + v) * 48 + 16 * t + mrow] = acc1[t][v];
    }
  }
  __syncthreads();

  // ---- phase 2: one lane = one pixel ----
  const float* wv = slot + lane * 48;  // 44 encoder channels for this pixel
  const int gp = waveBase + lane;      // global pixel index (b,h,w flattened)

  // m output: channels 0..7
  float* mo = out + (size_t)gp * 8;
  *(float4*)(mo)     = make_float4(wv[0], wv[1], wv[2], wv[3]);
  *(float4*)(mo + 4) = make_float4(wv[4], wv[5], wv[6], wv[7]);

  // scales: s = 0.001*(1-sig) + 1000*sig ; keep sqrt for S = (sqrt(s)R)^T (sqrt(s)R)
  float g[8];
#pragma unroll
  for (int c2 = 0; c2 < 8; ++c2) {
    float tv  = wv[8 + c2];
    float sig = 1.0f / (1.0f + __expf(-tv));
    float sp  = 0.001f + 999.999f * sig;
    g[c2] = __fsqrt_rn(sp);
  }

  // R = product of 28 Givens rotations applied to identity (all in registers)
  float R[8][8];
#pragma unroll
  for (int i = 0; i < 8; ++i)
#pragma unroll
    for (int j = 0; j < 8; ++j)
      R[i][j] = (i == j) ? 1.0f : 0.0f;

  {
    int c = 0;
#pragma unroll
    for (int i = 0; i < 7; ++i) {
#pragma unroll
      for (int j = i + 1; j < 8; ++j) {
        float ang = 3.14159265358979323846f * tanhf(wv[16 + c]);
        float cs = __cosf(ang);
        float ss = __sinf(ang);
#pragma unroll
        for (int r = 0; r < 8; ++r) {
          float ci = R[r][i], cj = R[r][j];
          R[r][i] = fmaf(ci, cs, cj * ss);
          R[r][j] = fmaf(cj, cs, -ci * ss);
        }
        ++c;
      }
    }
  }

  // scale rows: G = diag(sqrt(s)) * R
#pragma unroll
  for (int r = 0; r < 8; ++r)
#pragma unroll
    for (int k = 0; k < 8; ++k)
      R[r][k] *= g[r];

  // S = G^T G (exactly symmetric, so 0.5*(S+S^T) == S)
  float* So = out + M_ELEMS + (size_t)gp * 64;
#pragma unroll
  for (int i = 0; i < 8; ++i) {
    float sr[8];
#pragma unroll
    for (int k = 0; k < 8; ++k) {
      float s = 0.0f;
#pragma unroll
      for (int r = 0; r < 8; ++r) s = fmaf(R[r][i], R[r][k], s);
      sr[k] = s;
    }
    *(float4*)(So + i * 8)     = make_float4(sr[0], sr[1], sr[2], sr[3]);
    *(float4*)(So + i * 8 + 4) = make_float4(sr[4], sr[5], sr[6], sr[7]);
  }
}

extern "C" void kernel_launch(void* const* d_in, const int* in_sizes, int n_in,
                              void* d_out, int out_size, void* d_ws, size_t ws_size,
                              hipStream_t stream) {
  (void)in_sizes; (void)n_in; (void)d_ws; (void)ws_size; (void)out_size;
  const float* x    = (const float*)d_in[0];
  const float* Wenc = (const float*)d_in[1];
  const float* benc = (const float*)d_in[2];
  float* out        = (float*)d_out;
  // 262144 pixels / (32 pixels per wave * 8 waves per block) = 1024 blocks
  sepgpd_kernel<<<1024, 256, 0, stream>>>(x, Wenc, benc, out);
}